// AsymmetricWeightsDequantizer_54073638256714
// MI455X (gfx1250) — compile-verified
//
#include <hip/hip_runtime.h>

typedef __attribute__((ext_vector_type(2))) float v2f;
typedef __attribute__((ext_vector_type(8))) float v8f;

// Problem dims (fixed by the reference)
#define DIM_O   8192
#define DIM_I   8192
#define DIM_G   64
#define DIM_GS  128
#define DIM_K   128   // svd rank

// Tiling: one wave32 -> 16 rows x 64 cols of the output.
//   512 row-tiles * 128 col-strips = 65536 waves = 8192 blocks * 8 waves.
#define WAVES_PER_BLOCK 8
#define COL_STRIPS      (DIM_I / 64)   // 128

__global__ __launch_bounds__(256)
void dequant_lowrank_wmma(const int*   __restrict__ weight,   // [O, G*GS] int32
                          const float* __restrict__ scale,    // [O, G]
                          const float* __restrict__ zp,       // [O, G]
                          const float* __restrict__ up,       // [O, K]
                          const float* __restrict__ down,     // [K, I]
                          float*       __restrict__ out)      // [O, I]
{
    const int lane  = threadIdx.x & 31;
    const int wave  = threadIdx.x >> 5;
    const int wid   = blockIdx.x * WAVES_PER_BLOCK + wave;

    const int rtile  = wid >> 7;          // 0..511  (row block of 16)
    const int cstrip = wid & (COL_STRIPS - 1); // 0..127 (col block of 64)

    const int row0 = rtile * 16;
    const int col0 = cstrip * 64;
    const int grp  = col0 / DIM_GS;       // all 64 cols share one quant group
    const int half = lane >> 4;           // 0 | 1
    const int ln   = lane & 15;           // 0..15

    // ---- Accumulator init = dequantized weight (the addcmul part) ----
    // C/D f32 16x16 layout: VGPR v, lanes 0-15 -> M=v, lanes 16-31 -> M=v+8; N=lane%16.
    v8f acc[4];
#pragma unroll
    for (int v = 0; v < 8; ++v) {
        const int row = row0 + v + 8 * half;
        const float s = scale[row * DIM_G + grp];
        const float z = zp[row * DIM_G + grp];
        const int* wrow = weight + row * DIM_I + col0 + ln;
#pragma unroll
        for (int nt = 0; nt < 4; ++nt) {
            // weight is streamed exactly once: non-temporal so it doesn't
            // evict the L2-resident svd factors.
            const float w = (float)__builtin_nontemporal_load(wrow + nt * 16);
            acc[nt][v] = z + w * s;
        }
    }

    // ---- Low-rank correction: acc += up[16xK] @ down[Kx64] via f32 WMMA ----
    // A (16x4 f32) layout: lane<16 -> M=lane, {K=k0, k0+1}; lane>=16 -> {K=k0+2, k0+3}.
    // B (4x16 f32) layout mirrors A: row K striped across lanes, N=lane%16.
    const float* arow = up + (row0 + ln) * DIM_K;
#pragma unroll 4
    for (int k0 = 0; k0 < DIM_K; k0 += 4) {
        const int ka = k0 + 2 * half;     // even -> 8B aligned b64 load
        const v2f a = *(const v2f*)(arow + ka);
        const float* drow0 = down + ka * DIM_I + col0 + ln;
#pragma unroll
        for (int nt = 0; nt < 4; ++nt) {
            v2f b;
            b.x = drow0[nt * 16];
            b.y = drow0[DIM_I + nt * 16];
            acc[nt] = __builtin_amdgcn_wmma_f32_16x16x4_f32(
                /*neg_a=*/false, a, /*neg_b=*/false, b,
                /*c_mod=*/(short)0, acc[nt],
                /*reuse_a=*/false, /*reuse_b=*/false);
        }
    }

    // ---- Store (written once: non-temporal) ----
#pragma unroll
    for (int v = 0; v < 8; ++v) {
        const int row = row0 + v + 8 * half;
        float* orow = out + row * DIM_I + col0 + ln;
#pragma unroll
        for (int nt = 0; nt < 4; ++nt)
            __builtin_nontemporal_store(acc[nt][v], orow + nt * 16);
    }
}

extern "C" void kernel_launch(void* const* d_in, const int* in_sizes, int n_in,
                              void* d_out, int out_size, void* d_ws, size_t ws_size,
                              hipStream_t stream) {
    (void)in_sizes; (void)n_in; (void)out_size; (void)d_ws; (void)ws_size;
    const int*   weight = (const int*)  d_in[0];   // [O, G, GS] int32
    const float* scale  = (const float*)d_in[1];   // [O, G, 1]
    const float* zp     = (const float*)d_in[2];   // [O, G, 1]
    const float* up     = (const float*)d_in[3];   // [O, RANK]
    const float* down   = (const float*)d_in[4];   // [RANK, I]
    float*       out    = (float*)      d_out;     // [O, I]

    const int nWaves = (DIM_O / 16) * COL_STRIPS;          // 65536
    const int nBlocks = nWaves / WAVES_PER_BLOCK;          // 8192
    dequant_lowrank_wmma<<<nBlocks, WAVES_PER_BLOCK * 32, 0, stream>>>(
        weight, scale, zp, up, down, out);
}